// DWNBlock_25348896981260
// MI455X (gfx1250) — compile-verified
//
#include <hip/hip_runtime.h>
#include <math.h>

// ---------------------------------------------------------------------------
// LRU block for MI455X (gfx1250, wave32, WMMA, async global->LDS).
// fp32 throughout; GEMMs use V_WMMA_F32_16X16X4_F32 with double-buffered
// GLOBAL_LOAD_ASYNC_TO_LDS_B128 staging (ASYNCcnt pipeline).
// ---------------------------------------------------------------------------

typedef __attribute__((ext_vector_type(2))) float v2f;
typedef __attribute__((ext_vector_type(8))) float v8f;

#define D_MODEL 512
#define BATCH   8
#define SEQ     2048
#define NSTATE  256
#define MTOT    (BATCH * SEQ)          // 16384 rows
#define NCHUNK  16
#define CHLEN   (SEQ / NCHUNK)         // 128

// ---- workspace layout (floats) --------------------------------------------
#define OFF_Z     0u                          // [MTOT,512] layernorm output
#define OFF_BU    8388608u                    // [MTOT,512] Bu re||im -> states
#define OFF_Y     16777216u                   // [MTOT,512] Dpart -> y -> h
#define OFF_W1    25165824u                   // packed [512/4][1024][4]
#define OFF_W3    25690112u                   // packed [512/4][1024][4]
#define OFF_W2    26214400u                   // packed [512/4][512][4]
#define OFF_BIAS  26476544u                   // [1024] permuted bias
#define OFF_CARRY 26477568u                   // [8*16*512] scan carries

// ---------------------------------------------------------------------------
// CDNA5 async global->LDS copy + ASYNCcnt waits.
// Generic pointers into __shared__ have the LDS byte offset in their low
// 32 bits (flat LDS aperture truncation), which is what VDST expects.
// ---------------------------------------------------------------------------
__device__ __forceinline__ void async_copy_b128(const float* gptr, float* lptr)
{
  const unsigned int lds_off = (unsigned int)(unsigned long long)lptr;
  asm volatile("global_load_async_to_lds_b128 %0, %1, off"
               :: "v"(lds_off), "v"((unsigned long long)gptr)
               : "memory");
}

__device__ __forceinline__ void wait_async0()
{
#if __has_builtin(__builtin_amdgcn_s_wait_asynccnt)
  __builtin_amdgcn_s_wait_asynccnt(0);
#else
  asm volatile("s_wait_asynccnt 0x0" ::: "memory");
#endif
}

__device__ __forceinline__ void wait_async6()
{
#if __has_builtin(__builtin_amdgcn_s_wait_asynccnt)
  __builtin_amdgcn_s_wait_asynccnt(6);
#else
  asm volatile("s_wait_asynccnt 0x6" ::: "memory");
#endif
}

// ---------------------------------------------------------------------------
// LayerNorm: one block (128 threads) per row of 512.
// ---------------------------------------------------------------------------
__global__ __launch_bounds__(128) void ln_kernel(
    const float* __restrict__ x, const float* __restrict__ w,
    const float* __restrict__ b, float* __restrict__ z)
{
  const int m   = blockIdx.x;
  const int tid = threadIdx.x;
  const float4 v = ((const float4*)(x + (size_t)m * D_MODEL))[tid];
  float s1 = v.x + v.y + v.z + v.w;
  float s2 = v.x * v.x + v.y * v.y + v.z * v.z + v.w * v.w;
  #pragma unroll
  for (int off = 16; off > 0; off >>= 1) {
    s1 += __shfl_down(s1, off);
    s2 += __shfl_down(s2, off);
  }
  __shared__ float r1[4], r2[4];
  const int wv = tid >> 5, ln = tid & 31;
  if (ln == 0) { r1[wv] = s1; r2[wv] = s2; }
  __syncthreads();
  if (tid == 0) {
    float a = r1[0] + r1[1] + r1[2] + r1[3];
    float c = r2[0] + r2[1] + r2[2] + r2[3];
    float mu  = a * (1.0f / 512.0f);
    float var = c * (1.0f / 512.0f) - mu * mu;
    r1[0] = mu;
    r2[0] = rsqrtf(var + 1e-5f);
  }
  __syncthreads();
  const float mu = r1[0], rstd = r2[0];
  const float4 wv4 = ((const float4*)w)[tid];
  const float4 bv4 = ((const float4*)b)[tid];
  float4 o;
  o.x = (v.x - mu) * rstd * wv4.x + bv4.x;
  o.y = (v.y - mu) * rstd * wv4.y + bv4.y;
  o.z = (v.z - mu) * rstd * wv4.z + bv4.z;
  o.w = (v.w - mu) * rstd * wv4.w + bv4.w;
  ((float4*)(z + (size_t)m * D_MODEL))[tid] = o;
}

// ---------------------------------------------------------------------------
// Weight packing into [K/4][N][4] fragment-native layout.
// ---------------------------------------------------------------------------
__global__ __launch_bounds__(256) void pack_w1(
    const float* __restrict__ B_re, const float* __restrict__ B_im,
    const float* __restrict__ gamma_log, const float* __restrict__ D,
    float* __restrict__ Wp)
{
  const int id = blockIdx.x * 256 + threadIdx.x;   // 512*1024 threads
  const int k = id >> 10, n = id & 1023;
  float v;
  if (n < 256)      v = B_re[n * 512 + k] * __expf(gamma_log[n]);
  else if (n < 512) v = B_im[(n - 256) * 512 + k] * __expf(gamma_log[n - 256]);
  else              v = D[(n - 512) * 512 + k];
  Wp[(size_t)(k >> 2) * 4096 + n * 4 + (k & 3)] = v;
}

__global__ __launch_bounds__(256) void pack_w2(
    const float* __restrict__ C_re, const float* __restrict__ C_im,
    float* __restrict__ Wp)
{
  const int id = blockIdx.x * 256 + threadIdx.x;   // 512*512 threads
  const int k = id >> 9, n = id & 511;
  const float v = (k < 256) ? C_re[n * 256 + k] : -C_im[n * 256 + (k - 256)];
  Wp[(size_t)(k >> 2) * 2048 + n * 4 + (k & 3)] = v;
}

// W_out columns interleaved so (a_j, g_j) live in adjacent 16-wide subtiles
// of the same wave: col_a(j) = 32*(j/16)+(j%16), col_g(j) = col_a(j)+16.
__global__ __launch_bounds__(256) void pack_w3(
    const float* __restrict__ W_out, float* __restrict__ Wp)
{
  const int id = blockIdx.x * 256 + threadIdx.x;   // 512*1024 threads
  const int k = id >> 10, n = id & 1023;
  const int jj = (n >> 5) * 16 + (n & 15);
  const int srow = jj + ((n >> 4) & 1) * 512;
  Wp[(size_t)(k >> 2) * 4096 + n * 4 + (k & 3)] = W_out[srow * 512 + k];
}

__global__ __launch_bounds__(256) void pack_bias(
    const float* __restrict__ b_out, float* __restrict__ bp)
{
  const int n = blockIdx.x * 256 + threadIdx.x;    // 1024 threads
  const int jj = (n >> 5) * 16 + (n & 15);
  bp[n] = b_out[jj + ((n >> 4) & 1) * 512];
}

// ---------------------------------------------------------------------------
// WMMA GEMM with double-buffered async LDS staging.
// Block: 256 threads (8 waves), tile 64(M) x 128(N), K-step 32, K=512.
// Wave: 16x64 strip = 4 accumulators of v8f.
// MODE 1: out cols [0,512) -> out0 (Bu re||im); [512,1024) -> out1 (Dpart)
// MODE 2: out0[m,n] = gelu(out0[m,n] + acc)     (h = gelu(Dpart + C-proj))
// MODE 3: epilogue bias + GLU + residual -> out0
// ---------------------------------------------------------------------------
__device__ __forceinline__ void issue_tile(
    const float* __restrict__ A, const float* __restrict__ Wp, int N,
    int m0, int n0, int kb, int tid, float* sAb, float* sBb)
{
  // A tile: 64x32 floats, 2x b128 per thread
  #pragma unroll
  for (int i = 0; i < 2; i++) {
    const int idx = tid + i * 256;
    const int row = idx >> 3, c8 = idx & 7;
    const float* g = A + (size_t)(m0 + row) * 512 + kb + c8 * 4;
    async_copy_b128(g, sAb + row * 36 + c8 * 4);
  }
  // packed W tile: 8 panels x 128 n x 4 k, 4x b128 per thread
  #pragma unroll
  for (int i = 0; i < 4; i++) {
    const int idx = tid + i * 256;
    const int panel = idx >> 7, within = idx & 127;
    const float* g = Wp + ((size_t)((kb >> 2) + panel) * N + n0 + within) * 4;
    async_copy_b128(g, sBb + (panel * 128 + within) * 4);
  }
}

__device__ __forceinline__ void compute_tile(
    const float* sAb, const float* sBb, v8f acc[4],
    int wm, int wn, int laneM, int laneH)
{
  #pragma unroll
  for (int kk4 = 0; kk4 < 8; kk4++) {
    const float* ap = sAb + (wm * 16 + laneM) * 36 + kk4 * 4 + 2 * laneH;
    v2f a; a.x = ap[0]; a.y = ap[1];
    #pragma unroll
    for (int j = 0; j < 4; j++) {
      const float* bp = sBb + kk4 * 512 + (wn * 64 + j * 16 + laneM) * 4 + 2 * laneH;
      v2f b; b.x = bp[0]; b.y = bp[1];
      acc[j] = __builtin_amdgcn_wmma_f32_16x16x4_f32(
          false, a, false, b, (short)0, acc[j], false, false);
    }
  }
}

template <int MODE>
__global__ __launch_bounds__(256) void gemm_wmma_kernel(
    const float* __restrict__ A, const float* __restrict__ Wp,
    const float* __restrict__ biasp, const float* __restrict__ X,
    float* __restrict__ out0, float* __restrict__ out1, int N)
{
  __shared__ __align__(16) float sA[2][64 * 36];
  __shared__ __align__(16) float sB[2][8 * 128 * 4];

  const int tid   = threadIdx.x;
  const int lane  = tid & 31;
  const int wave  = tid >> 5;
  const int laneM = lane & 15;
  const int laneH = lane >> 4;
  const int wm    = wave & 3;   // 4 M-waves
  const int wn    = wave >> 2;  // 2 N-waves
  const int m0    = blockIdx.x * 64;
  const int n0    = blockIdx.y * 128;

  v8f acc[4];
  #pragma unroll
  for (int j = 0; j < 4; j++)
    #pragma unroll
    for (int r = 0; r < 8; r++) acc[j][r] = 0.0f;

  issue_tile(A, Wp, N, m0, n0, 0, tid, sA[0], sB[0]);

  #pragma unroll 1
  for (int kb = 0; kb < 512; kb += 64) {
    issue_tile(A, Wp, N, m0, n0, kb + 32, tid, sA[1], sB[1]);
    wait_async6();                 // buffer-0 tile landed (in-order ASYNCcnt)
    __syncthreads();
    compute_tile(sA[0], sB[0], acc, wm, wn, laneM, laneH);
    __syncthreads();               // all waves done reading buffer 0
    if (kb + 64 < 512) {
      issue_tile(A, Wp, N, m0, n0, kb + 64, tid, sA[0], sB[0]);
      wait_async6();               // buffer-1 tile landed
    } else {
      wait_async0();
    }
    __syncthreads();
    compute_tile(sA[1], sB[1], acc, wm, wn, laneM, laneH);
    __syncthreads();               // all waves done reading buffer 1
  }

  const int mrow = m0 + wm * 16;
  if (MODE == 1) {
    #pragma unroll
    for (int j = 0; j < 4; j++) {
      const int n = n0 + wn * 64 + j * 16 + laneM;
      #pragma unroll
      for (int r = 0; r < 8; r++) {
        const int m = mrow + r + 8 * laneH;
        const float v = acc[j][r];
        if (n < 512) out0[(size_t)m * 512 + n] = v;
        else         out1[(size_t)m * 512 + (n - 512)] = v;
      }
    }
  } else if (MODE == 2) {
    #pragma unroll
    for (int j = 0; j < 4; j++) {
      const int n = n0 + wn * 64 + j * 16 + laneM;
      #pragma unroll
      for (int r = 0; r < 8; r++) {
        const int m = mrow + r + 8 * laneH;
        const float yv = out0[(size_t)m * 512 + n] + acc[j][r];
        out0[(size_t)m * 512 + n] =
            0.5f * yv * (1.0f + erff(yv * 0.70710678118f));   // h = gelu(y)
      }
    }
  } else {
    #pragma unroll
    for (int p = 0; p < 2; p++) {
      const int colg = n0 + wn * 64 + p * 32 + laneM;     // a-column index
      const int jj   = (colg >> 5) * 16 + (colg & 15);    // logical output col
      const float ba = biasp[colg];
      const float bg = biasp[colg + 16];
      #pragma unroll
      for (int r = 0; r < 8; r++) {
        const int m = mrow + r + 8 * laneH;
        const float av = acc[2 * p][r] + ba;
        const float gv = acc[2 * p + 1][r] + bg;
        const float sg = 1.0f / (1.0f + __expf(-gv));
        out0[(size_t)m * 512 + jj] = av * sg + X[(size_t)m * 512 + jj];
      }
    }
  }
}

// ---------------------------------------------------------------------------
// Chunked associative scan of x_t = lam*x_{t-1} + Bu_t (complex, diagonal).
// Bu layout: [MTOT, 512], cols [0,256)=re, [256,512)=im. In-place.
// ---------------------------------------------------------------------------
__device__ __forceinline__ void lam_of(const float* nu_log, const float* th_log,
                                       int s, float& lr, float& li)
{
  const float mag = __expf(-__expf(nu_log[s]));
  const float th  = __expf(th_log[s]);
  lr = mag * __cosf(th);
  li = mag * __sinf(th);
}

__global__ __launch_bounds__(256) void scan_local(
    const float* __restrict__ nu_log, const float* __restrict__ th_log,
    float* __restrict__ bu, float* __restrict__ carry)
{
  const int s = threadIdx.x;          // 256 states
  const int b = blockIdx.x;           // 8 batch
  const int c = blockIdx.y;           // 16 chunks
  float lr, li; lam_of(nu_log, th_log, s, lr, li);
  float sr = 0.0f, si = 0.0f;
  size_t base = (size_t)(b * SEQ + c * CHLEN) * 512 + s;
  for (int i = 0; i < CHLEN; i++) {
    const float br = bu[base], bi = bu[base + 256];
    const float nr = lr * sr - li * si + br;
    const float ni = lr * si + li * sr + bi;
    bu[base] = nr; bu[base + 256] = ni;
    sr = nr; si = ni;
    base += 512;
  }
  const size_t ci = (size_t)(b * NCHUNK + c) * 512 + s;
  carry[ci] = sr; carry[ci + 256] = si;
}

__global__ __launch_bounds__(256) void scan_carry(
    const float* __restrict__ nu_log, const float* __restrict__ th_log,
    float* __restrict__ carry)
{
  const int s = threadIdx.x;
  const int b = blockIdx.x;
  float lr, li; lam_of(nu_log, th_log, s, lr, li);
  float pr = lr, pi = li;               // lam^128 by 7 squarings
  #pragma unroll
  for (int i = 0; i < 7; i++) {
    const float nr = pr * pr - pi * pi;
    const float ni = 2.0f * pr * pi;
    pr = nr; pi = ni;
  }
  float Xr = 0.0f, Xi = 0.0f;           // true state at end of prior chunks
  for (int c = 0; c < NCHUNK; c++) {
    const size_t ci = (size_t)(b * NCHUNK + c) * 512 + s;
    const float cr = carry[ci], cim = carry[ci + 256];
    carry[ci] = Xr; carry[ci + 256] = Xi;        // overwrite with prefix
    const float nXr = pr * Xr - pi * Xi + cr;
    const float nXi = pr * Xi + pi * Xr + cim;
    Xr = nXr; Xi = nXi;
  }
}

__global__ __launch_bounds__(256) void scan_fixup(
    const float* __restrict__ nu_log, const float* __restrict__ th_log,
    float* __restrict__ bu, const float* __restrict__ carry)
{
  const int s = threadIdx.x;
  const int b = blockIdx.x;
  const int c = blockIdx.y;
  const size_t ci = (size_t)(b * NCHUNK + c) * 512 + s;
  const float Xr = carry[ci], Xi = carry[ci + 256];
  if (Xr == 0.0f && Xi == 0.0f) return;          // chunk 0 and exact zeros
  float lr, li; lam_of(nu_log, th_log, s, lr, li);
  float fr = lr, fi = li;                        // lam^(i+1)
  size_t base = (size_t)(b * SEQ + c * CHLEN) * 512 + s;
  for (int i = 0; i < CHLEN; i++) {
    bu[base]       += fr * Xr - fi * Xi;
    bu[base + 256] += fr * Xi + fi * Xr;
    const float nfr = fr * lr - fi * li;
    const float nfi = fr * li + fi * lr;
    fr = nfr; fi = nfi;
    base += 512;
  }
}

// ---------------------------------------------------------------------------
extern "C" void kernel_launch(void* const* d_in, const int* in_sizes, int n_in,
                              void* d_out, int out_size, void* d_ws, size_t ws_size,
                              hipStream_t stream)
{
  const float* x         = (const float*)d_in[0];
  const float* ln_w      = (const float*)d_in[1];
  const float* ln_b      = (const float*)d_in[2];
  const float* nu_log    = (const float*)d_in[3];
  const float* theta_log = (const float*)d_in[4];
  const float* gamma_log = (const float*)d_in[5];
  const float* B_re      = (const float*)d_in[6];
  const float* B_im      = (const float*)d_in[7];
  const float* C_re      = (const float*)d_in[8];
  const float* C_im      = (const float*)d_in[9];
  const float* Dm        = (const float*)d_in[10];
  const float* W_out     = (const float*)d_in[11];
  const float* b_out     = (const float*)d_in[12];

  float* ws    = (float*)d_ws;
  float* z     = ws + OFF_Z;
  float* bu    = ws + OFF_BU;
  float* ybuf  = ws + OFF_Y;
  float* w1    = ws + OFF_W1;
  float* w2    = ws + OFF_W2;
  float* w3    = ws + OFF_W3;
  float* biasp = ws + OFF_BIAS;
  float* carry = ws + OFF_CARRY;
  float* out   = (float*)d_out;

  // LayerNorm + weight packing (independent, all on stream)
  ln_kernel<<<MTOT, 128, 0, stream>>>(x, ln_w, ln_b, z);
  pack_w1<<<2048, 256, 0, stream>>>(B_re, B_im, gamma_log, Dm, w1);
  pack_w2<<<1024, 256, 0, stream>>>(C_re, C_im, w2);
  pack_w3<<<2048, 256, 0, stream>>>(W_out, w3);
  pack_bias<<<4, 256, 0, stream>>>(b_out, biasp);

  // GEMM1: z @ [B_norm_re^T | B_norm_im^T | D^T]  -> Bu (re||im), Dpart (in ybuf)
  gemm_wmma_kernel<1><<<dim3(MTOT / 64, 8), 256, 0, stream>>>(
      z, w1, nullptr, nullptr, bu, ybuf, 1024);

  // Chunked complex scan over L (in place on bu)
  scan_local<<<dim3(BATCH, NCHUNK), 256, 0, stream>>>(nu_log, theta_log, bu, carry);
  scan_carry<<<BATCH, 256, 0, stream>>>(nu_log, theta_log, carry);
  scan_fixup<<<dim3(BATCH, NCHUNK), 256, 0, stream>>>(nu_log, theta_log, bu, carry);

  // GEMM2: states @ [C_re^T ; -C_im^T] + Dpart, fused GELU -> h (in ybuf)
  gemm_wmma_kernel<2><<<dim3(MTOT / 64, 4), 256, 0, stream>>>(
      bu, w2, nullptr, nullptr, ybuf, nullptr, 512);

  // GEMM3: h @ W_out^T (+bias), GLU, +residual -> out
  gemm_wmma_kernel<3><<<dim3(MTOT / 64, 8), 256, 0, stream>>>(
      ybuf, w3, biasp, x, out, nullptr, 1024);
}